// QuantizationLayer_446676598908
// MI455X (gfx1250) — compile-verified
//
#include <hip/hip_runtime.h>

// Quantize 16x1M xy points to a 128x128 grid and scatter-add a per-batch
// histogram. Output layout (int32, concatenated in reference return order):
//   d_out[0 .. 33554432)              : q  [B, N, 2]  (qx, qy interleaved)
//   d_out[33554432 .. 33816576)       : vox [B, 128, 128]
//
// Strategy (MI455X / gfx1250):
//  - Streaming phase is HBM-bound (256 MB @ 23.3 TB/s ~= 11 us): use b128
//    non-temporal loads/stores so the stream does not thrash L2.
//  - Histogram privatized in LDS (64 KB per block = one batch's grid;
//    320 KB/WGP -> up to 5 resident blocks/WGP). 16M ds_add_u32 LDS atomics
//    instead of 16M L2 RMWs.
//  - 32 blocks per batch -> flush is ~8.4M (minus ~13% zero bins) no-return
//    global_atomic_add_u32, order-independent => deterministic.

typedef float __attribute__((ext_vector_type(4))) f32x4;
typedef int   __attribute__((ext_vector_type(4))) i32x4;

constexpr int   GRID_W = 128;
constexpr int   GRID_H = 128;
constexpr int   BATCH  = 16;
constexpr int   NPTS   = 1 << 20;            // points per batch
constexpr int   BPB    = 32;                 // blocks per batch
constexpr int   BLOCK  = 256;                // threads per block (8 wave32)
constexpr int   BINS   = GRID_W * GRID_H;    // 16384 bins = 64 KB LDS
constexpr int   PTS_PER_BLOCK = NPTS / BPB;          // 32768
constexpr int   VEC_PER_BLOCK = PTS_PER_BLOCK / 2;   // 16384 float4 (2 pts each)
constexpr int   ITERS  = VEC_PER_BLOCK / BLOCK;      // 64
constexpr float SCALE  = 127.0f;             // min(W,H) - 1

__global__ __launch_bounds__(BLOCK)
void vox_zero_kernel(i32x4* __restrict__ vox4) {
    int i = blockIdx.x * BLOCK + threadIdx.x;
    i32x4 z = {0, 0, 0, 0};
    vox4[i] = z;
}

__global__ __launch_bounds__(BLOCK)
void quant_hist_kernel(const f32x4* __restrict__ xy4,
                       i32x4* __restrict__ q4,
                       int* __restrict__ vox) {
    __shared__ int hist[BINS];

    const int tid = threadIdx.x;

    // Zero the per-block LDS histogram (64 ints per thread).
    #pragma unroll
    for (int i = tid; i < BINS; i += BLOCK) hist[i] = 0;
    __syncthreads();

    const int b    = blockIdx.x / BPB;   // batch this block serves
    const int sub  = blockIdx.x % BPB;   // chunk within the batch
    const long base4 = (long)b * (NPTS / 2) + (long)sub * VEC_PER_BLOCK;

    // Main streaming loop: each f32x4 = two (x,y) points.
    // Non-temporal b128 load -> quantize -> non-temporal b128 store of q,
    // plus two LDS atomic increments. Unroll x4 for outstanding-load depth
    // (only ~8 waves/SIMD-pair at 64 KB LDS occupancy).
    #pragma unroll 4
    for (int it = 0; it < ITERS; ++it) {
        const long idx = base4 + (long)it * BLOCK + tid;

        f32x4 v = __builtin_nontemporal_load(&xy4[idx]);

        const int qx0 = (int)(v.x * SCALE);   // v_cvt_i32_f32: trunc toward 0
        const int qy0 = (int)(v.y * SCALE);
        const int qx1 = (int)(v.z * SCALE);
        const int qy1 = (int)(v.w * SCALE);

        i32x4 qv;
        qv.x = qx0; qv.y = qy0; qv.z = qx1; qv.w = qy1;
        __builtin_nontemporal_store(qv, &q4[idx]);

        atomicAdd(&hist[qy0 * GRID_W + qx0], 1);   // ds_add_u32 (no return)
        atomicAdd(&hist[qy1 * GRID_W + qx1], 1);
    }
    __syncthreads();

    // Merge private histogram into the global grid. Avg 2 counts/bin in this
    // block's chunk -> skip ~13% empty bins. No-return global atomic add.
    int* __restrict__ voxb = vox + b * BINS;
    #pragma unroll
    for (int i = tid; i < BINS; i += BLOCK) {
        const int c = hist[i];
        if (c) atomicAdd(&voxb[i], c);
    }
}

extern "C" void kernel_launch(void* const* d_in, const int* in_sizes, int n_in,
                              void* d_out, int out_size, void* d_ws, size_t ws_size,
                              hipStream_t stream) {
    (void)in_sizes; (void)n_in; (void)out_size; (void)d_ws; (void)ws_size;

    const f32x4* xy4 = (const f32x4*)d_in[0];
    int*   out  = (int*)d_out;
    i32x4* q4   = (i32x4*)out;                                 // 32M ints
    int*   vox  = out + (size_t)BATCH * NPTS * 2;              // 262144 ints

    // 1) Zero the vox region (harness poisons d_out; atomics need zeros).
    //    262144 ints / 4 per thread / 256 threads = 256 blocks.
    vox_zero_kernel<<<(BATCH * BINS / 4) / BLOCK, BLOCK, 0, stream>>>((i32x4*)vox);

    // 2) Fused quantize + privatized histogram. 16 batches x 32 blocks.
    quant_hist_kernel<<<BATCH * BPB, BLOCK, 0, stream>>>(xy4, q4, vox);
}